// GCN_9783935500634
// MI455X (gfx1250) — compile-verified
//
#include <hip/hip_runtime.h>

typedef __attribute__((ext_vector_type(2))) float v2f;
typedef __attribute__((ext_vector_type(8))) float v8f;
typedef int v4i_ __attribute__((vector_size(4 * sizeof(int))));

// ---------------------------------------------------------------------------
// CDNA5 async global->LDS staging (ASYNCcnt path), with portable fallback.
// Probe result (round 2 diagnostic): builtin exists; param0 is
//   'int __attribute__((vector_size(16))) addrspace(1)*' (non-const).
// ---------------------------------------------------------------------------
#if defined(__HIP_DEVICE_COMPILE__) && \
    __has_builtin(__builtin_amdgcn_global_load_async_to_lds_b128) && \
    __has_builtin(__builtin_amdgcn_s_wait_asynccnt)
#define GCN_ASYNC_LDS 1
#endif

__device__ __forceinline__ void copy16_to_lds(void* lds, const void* glb) {
#ifdef GCN_ASYNC_LDS
    __builtin_amdgcn_global_load_async_to_lds_b128(
        (__attribute__((address_space(1))) v4i_*)(uintptr_t)glb,
        (__attribute__((address_space(3))) v4i_*)(uintptr_t)lds, 0, 0);
#else
    *(float4*)lds = *(const float4*)glb;
#endif
}

__device__ __forceinline__ void wait_stage_le2() {
#ifdef GCN_ASYNC_LDS
    __builtin_amdgcn_s_wait_asynccnt(2);   // allow the prefetched stage in flight
#endif
}
__device__ __forceinline__ void wait_stage_le0() {
#ifdef GCN_ASYNC_LDS
    __builtin_amdgcn_s_wait_asynccnt(0);
#endif
}

// ---------------------------------------------------------------------------
// Degree / normalization
// ---------------------------------------------------------------------------
__global__ void deg_init_kernel(float* deg, int n) {
    int i = blockIdx.x * blockDim.x + threadIdx.x;
    if (i < n) deg[i] = 1.0f;               // self-loop contribution
}

__global__ void deg_accum_kernel(const long long* __restrict__ dst, float* deg, int e) {
    int i = blockIdx.x * blockDim.x + threadIdx.x;
    if (i < e) atomicAdd(&deg[(int)dst[i]], 1.0f);
}

__global__ void deg_rsqrt_kernel(float* deg, int n) {
    int i = blockIdx.x * blockDim.x + threadIdx.x;
    if (i < n) deg[i] = rsqrtf(deg[i]);     // deg >= 1 always (self loops)
}

// ---------------------------------------------------------------------------
// Tiled fp32 WMMA GEMM:  C[M,N] = A[M,K] @ B[K,N]          (K <= 512, 16 | K)
// Workgroup = 256 threads (8 wave32), tile = 128(M) x 64(N).
//  - Whole Kx64 B panel staged once in LDS, column-major (stride 516: b64
//    fragment loads, conflict-free banks) -> barrier-free main loop.
//  - Each wave double-buffers its private 16x16 A sub-tile via async
//    global->LDS b128 copies (ASYNCcnt), overlapping DMA with WMMA.
// ---------------------------------------------------------------------------
#define BSTR 516   // B column stride (floats): 516*i mod 64 = 4i -> 16 distinct banks
#define ASTR 28    // A row stride (floats): 112B = 7*16 (b128-aligned), 28*i mod 64 distinct

__global__ __launch_bounds__(256) void gemm_wmma_f32(
    const float* __restrict__ A, const float* __restrict__ B,
    float* __restrict__ C, int M, int K, int N)
{
    __shared__ __align__(16) float Bs[64 * BSTR];            // 132,096 B
    __shared__ __align__(16) float As[8 * 2 * 16 * ASTR];    //  28,672 B

    const int tid   = threadIdx.x;
    const int wave  = tid >> 5;
    const int lane  = tid & 31;
    const int m_blk = blockIdx.x * 128;
    const int n_blk = blockIdx.y * 64;

    // ---- stage full B panel (K x 64), row-major global -> column-major LDS ----
    for (int idx = tid; idx < (K << 4); idx += 256) {
        const int c4 = (idx & 15) << 2;
        const int k  = idx >> 4;
        const float4 v = *(const float4*)(B + (size_t)k * N + n_blk + c4);
        Bs[(c4 + 0) * BSTR + k] = v.x;
        Bs[(c4 + 1) * BSTR + k] = v.y;
        Bs[(c4 + 2) * BSTR + k] = v.z;
        Bs[(c4 + 3) * BSTR + k] = v.w;
    }
    __syncthreads();    // only workgroup-wide sync in the kernel

    float* Aw = &As[wave * (2 * 16 * ASTR)];   // this wave's double buffer

    // stage a 16x16 A sub-tile for this wave: 64 b128 chunks, 2 per lane
    auto stage = [&](int buf, int k0) {
        #pragma unroll
        for (int j = 0; j < 2; ++j) {
            const int chunk = lane + (j << 5);   // 0..63
            const int row   = chunk >> 2;        // 0..15
            const int c4    = (chunk & 3) << 2;  // 0,4,8,12
            int grow = m_blk + (wave << 4) + row;
            if (grow >= M) grow = M - 1;         // clamp: rows never stored
            copy16_to_lds(&Aw[buf * (16 * ASTR) + row * ASTR + c4],
                          A + (size_t)grow * K + k0 + c4);
        }
    };

    v8f acc[4] = {};
    const int arow = lane & 15;                 // A-frag row (M within tile)
    const int kb   = (lane >> 4) << 1;          // k sub-offset: 0 or 2
    const int ncol = lane & 15;                 // column within 16x16 n-tile

    int buf = 0;
    stage(0, 0);
    for (int k0 = 0; k0 < K; k0 += 16) {
        const bool pf = (k0 + 16) < K;
        if (pf) { stage(buf ^ 1, k0 + 16); wait_stage_le2(); }
        else    { wait_stage_le0(); }

        const float* Ab = &Aw[buf * (16 * ASTR)];
        #pragma unroll
        for (int kk = 0; kk < 16; kk += 4) {
            const float* ap = &Ab[arow * ASTR + kk + kb];
            v2f a; a[0] = ap[0]; a[1] = ap[1];          // contiguous -> ds_load_b64
            #pragma unroll
            for (int n = 0; n < 4; ++n) {
                const float* bp = &Bs[((n << 4) + ncol) * BSTR + k0 + kk + kb];
                v2f b; b[0] = bp[0]; b[1] = bp[1];      // contiguous -> ds_load_b64
                acc[n] = __builtin_amdgcn_wmma_f32_16x16x4_f32(
                    false, a, false, b, (short)0, acc[n], false, false);
            }
        }
        buf ^= 1;
    }

    // ---- store: C/D layout -> row = r + (lane/16)*8, col = lane%16 ----
    const int rbase = m_blk + (wave << 4) + ((lane >> 4) << 3);
    #pragma unroll
    for (int n = 0; n < 4; ++n) {
        #pragma unroll
        for (int r = 0; r < 8; ++r) {
            const int row = rbase + r;
            if (row < M)
                C[(size_t)row * N + n_blk + (n << 4) + ncol] = acc[n][r];
        }
    }
}

// ---------------------------------------------------------------------------
// Aggregation: out = self-loop term, then scatter-add edge messages
// ---------------------------------------------------------------------------
__global__ void agg_self_kernel(const float* __restrict__ lin,
                                const float* __restrict__ dinv,
                                float* __restrict__ out,
                                long long total_vec4, int node_shift)
{
    for (long long i = (long long)blockIdx.x * blockDim.x + threadIdx.x;
         i < total_vec4;
         i += (long long)gridDim.x * blockDim.x) {
        const int node = (int)(i >> node_shift);
        float s = dinv[node];
        s = s * s;                                   // self-loop norm = 1/deg
        float4 v = ((const float4*)lin)[i];
        ((float4*)out)[i] = make_float4(v.x * s, v.y * s, v.z * s, v.w * s);
    }
}

__global__ __launch_bounds__(256) void agg_edges_kernel(
    const float* __restrict__ lin,
    const long long* __restrict__ src, const long long* __restrict__ dst,
    const float* __restrict__ dinv,
    float* __restrict__ out, long long E, int F)
{
    const long long wid = (long long)blockIdx.x * 8 + (threadIdx.x >> 5);
    if (wid >= E) return;
    const int lane = threadIdx.x & 31;
    const int s = (int)src[wid];
    const int d = (int)dst[wid];
    const float norm = dinv[s] * dinv[d];
    const float4* ls = (const float4*)(lin + (size_t)s * F);
    float* od = out + (size_t)d * F;
    const int nv = F >> 2;
    for (int i = lane; i < nv; i += 32) {
        float4 v = ls[i];
        atomicAdd(od + 4 * i + 0, v.x * norm);
        atomicAdd(od + 4 * i + 1, v.y * norm);
        atomicAdd(od + 4 * i + 2, v.z * norm);
        atomicAdd(od + 4 * i + 3, v.w * norm);
    }
}

__global__ void bias_act_kernel(float* __restrict__ h, const float* __restrict__ b,
                                long long total, int fmask, int relu)
{
    for (long long i = (long long)blockIdx.x * blockDim.x + threadIdx.x;
         i < total;
         i += (long long)gridDim.x * blockDim.x) {
        float v = h[i] + b[(int)i & fmask];
        if (relu) v = fmaxf(v, 0.0f);
        h[i] = v;
    }
}

// ---------------------------------------------------------------------------
// Readout: per-graph mean+max pooling, then pooled @ Wm + bm
// ---------------------------------------------------------------------------
__device__ inline void atomicMaxFloat(float* addr, float v) {
    if (v >= 0.0f) atomicMax((int*)addr, __float_as_int(v));
    else           atomicMin((unsigned int*)addr, __float_as_uint(v));
}

__global__ void readout_init_kernel(float* sums, float* cnt, float* mx, int gf, int g) {
    int i = blockIdx.x * blockDim.x + threadIdx.x;
    if (i < gf) { sums[i] = 0.0f; mx[i] = __int_as_float(0xFF800000); }  // -inf
    if (i < g) cnt[i] = 0.0f;
}

__global__ __launch_bounds__(256) void readout_accum_kernel(
    const float* __restrict__ h, const long long* __restrict__ batch,
    float* sums, float* cnt, float* mx, int n, int F)
{
    const int wid = blockIdx.x * 8 + (threadIdx.x >> 5);
    if (wid >= n) return;
    const int lane = threadIdx.x & 31;
    const int g = (int)batch[wid];
    const float* row = h + (size_t)wid * F;
    float* sg = sums + (size_t)g * F;
    float* mg = mx + (size_t)g * F;
    for (int f = lane; f < F; f += 32) {
        const float v = row[f];
        atomicAdd(sg + f, v);
        atomicMaxFloat(mg + f, v);
    }
    if (lane == 0) atomicAdd(cnt + g, 1.0f);
}

__global__ void final_mlp_kernel(const float* __restrict__ sums,
                                 const float* __restrict__ cnt,
                                 const float* __restrict__ mx,
                                 const float* __restrict__ Wm,
                                 const float* __restrict__ bm,
                                 float* __restrict__ out, int G, int F3)
{
    const int idx = blockIdx.x * blockDim.x + threadIdx.x;
    if (idx >= G * F3) return;
    const int g = idx / F3;
    const int j = idx - g * F3;
    const float c = cnt[g];
    const float invc = 1.0f / fmaxf(c, 1.0f);
    const bool has = c > 0.0f;
    float acc = bm[j];
    const float* sg = sums + (size_t)g * F3;
    const float* mg = mx + (size_t)g * F3;
    for (int k = 0; k < F3; ++k) {
        acc += (sg[k] * invc) * Wm[(size_t)k * F3 + j];
        const float mv = has ? mg[k] : 0.0f;
        acc += mv * Wm[(size_t)(F3 + k) * F3 + j];
    }
    out[idx] = acc;
}

// ---------------------------------------------------------------------------
// Launch: 3x (WMMA GEMM -> self-loop -> edge scatter -> bias/relu), readout
// ---------------------------------------------------------------------------
extern "C" void kernel_launch(void* const* d_in, const int* in_sizes, int n_in,
                              void* d_out, int out_size, void* d_ws, size_t ws_size,
                              hipStream_t stream)
{
    const float*     x     = (const float*)d_in[0];
    const long long* adj   = (const long long*)d_in[1];
    const long long* batch = (const long long*)d_in[2];
    const float* W1 = (const float*)d_in[3];  const float* b1 = (const float*)d_in[4];
    const float* W2 = (const float*)d_in[5];  const float* b2 = (const float*)d_in[6];
    const float* W3 = (const float*)d_in[7];  const float* b3 = (const float*)d_in[8];
    const float* Wm = (const float*)d_in[9];  const float* bm = (const float*)d_in[10];
    float* out = (float*)d_out;

    const int       Nn  = in_sizes[2];
    const long long Ee  = (long long)in_sizes[1] / 2;
    const int       F1  = in_sizes[4];
    const int       F2  = in_sizes[6];
    const int       F3  = in_sizes[8];
    const int       FIN = in_sizes[3] / F1;
    const int       Gg  = out_size / F3;

    const long long* srcp = adj;
    const long long* dstp = adj + Ee;

    // workspace layout (floats)
    float* ws   = (float*)d_ws;
    float* dinv = ws;
    float* sums = dinv + Nn;
    float* cnt  = sums + (size_t)Gg * F3;
    float* mx   = cnt + Gg;
    size_t hdr  = (size_t)Nn + (size_t)Gg * F3 + (size_t)Gg + (size_t)Gg * F3;
    hdr = (hdr + 3) & ~(size_t)3;                // float4 alignment
    float* R0 = ws + hdr;
    float* R1 = R0 + (size_t)Nn * FIN;
    float* R2 = R1 + (size_t)Nn * FIN;

    const int T = 256;

    // --- degrees & symmetric normalization ---
    deg_init_kernel<<<(Nn + T - 1) / T, T, 0, stream>>>(dinv, Nn);
    deg_accum_kernel<<<(int)((Ee + T - 1) / T), T, 0, stream>>>(dstp, dinv, (int)Ee);
    deg_rsqrt_kernel<<<(Nn + T - 1) / T, T, 0, stream>>>(dinv, Nn);

    const int edge_blocks = (int)((Ee + 7) / 8);

    // --- layer 1: x[N,FIN] @ W1 -> R1; agg -> R2; +b1, relu ---
    {
        dim3 grid((Nn + 127) / 128, F1 / 64);
        gemm_wmma_f32<<<grid, T, 0, stream>>>(x, W1, R1, Nn, FIN, F1);
    }
    {
        long long tv = (long long)Nn * (F1 >> 2);
        int shift = __builtin_ctz(F1 >> 2);
        agg_self_kernel<<<8192, T, 0, stream>>>(R1, dinv, R2, tv, shift);
        agg_edges_kernel<<<edge_blocks, T, 0, stream>>>(R1, srcp, dstp, dinv, R2, Ee, F1);
        bias_act_kernel<<<8192, T, 0, stream>>>(R2, b1, (long long)Nn * F1, F1 - 1, 1);
    }

    // --- layer 2: R2[N,F1] @ W2 -> R1; agg -> R0; +b2, relu ---
    {
        dim3 grid((Nn + 127) / 128, F2 / 64);
        gemm_wmma_f32<<<grid, T, 0, stream>>>(R2, W2, R1, Nn, F1, F2);
    }
    {
        long long tv = (long long)Nn * (F2 >> 2);
        int shift = __builtin_ctz(F2 >> 2);
        agg_self_kernel<<<8192, T, 0, stream>>>(R1, dinv, R0, tv, shift);
        agg_edges_kernel<<<edge_blocks, T, 0, stream>>>(R1, srcp, dstp, dinv, R0, Ee, F2);
        bias_act_kernel<<<8192, T, 0, stream>>>(R0, b2, (long long)Nn * F2, F2 - 1, 1);
    }

    // --- layer 3: R0[N,F2] @ W3 -> R1; agg -> R2; +b3 (no relu) ---
    {
        dim3 grid((Nn + 127) / 128, F3 / 64);
        gemm_wmma_f32<<<grid, T, 0, stream>>>(R0, W3, R1, Nn, F2, F3);
    }
    {
        long long tv = (long long)Nn * (F3 >> 2);
        int shift = __builtin_ctz(F3 >> 2);
        agg_self_kernel<<<8192, T, 0, stream>>>(R1, dinv, R2, tv, shift);
        agg_edges_kernel<<<edge_blocks, T, 0, stream>>>(R1, srcp, dstp, dinv, R2, Ee, F3);
        bias_act_kernel<<<8192, T, 0, stream>>>(R2, b3, (long long)Nn * F3, F3 - 1, 0);
    }

    // --- readout: mean+max pool per graph, then pooled @ Wm + bm ---
    readout_init_kernel<<<(Gg * F3 + T - 1) / T, T, 0, stream>>>(sums, cnt, mx, Gg * F3, Gg);
    readout_accum_kernel<<<(Nn + 7) / 8, T, 0, stream>>>(R2, batch, sums, cnt, mx, Nn, F3);
    final_mlp_kernel<<<(Gg * F3 + T - 1) / T, T, 0, stream>>>(sums, cnt, mx, Wm, bm, out, Gg, F3);
}